// FineMatching_66829691126061
// MI455X (gfx1250) — compile-verified
//
#include <hip/hip_runtime.h>
#include <hip/hip_bf16.h>

typedef __attribute__((ext_vector_type(2))) float v2f;
typedef __attribute__((ext_vector_type(8))) float v8f;

#define WWIN   8
#define WPAD   10
#define NL     64      // l positions (8x8)
#define NR     100     // r positions (10x10)
#define CCH    128     // channels
#define CFINE  120     // channels used by conf_f
#define FS     132     // padded LDS stride for feature rows (132 % 64 == 4 -> conflict free col reads)
#define CS     101     // padded LDS stride for conf rows (odd -> conflict free col reads)
#define NTILE  7       // 7 x 16 = 112 >= 100 columns

// async global->LDS builtin takes (int4 addrspace(1)*, int4 addrspace(3)*, imm, imm)
typedef int int4vs __attribute__((vector_size(16)));
typedef __attribute__((address_space(1))) int4vs g_int4;
typedef __attribute__((address_space(3))) int4vs l_int4;

__device__ __forceinline__ void wait_asynccnt0() {
#if __has_builtin(__builtin_amdgcn_s_wait_asynccnt)
    __builtin_amdgcn_s_wait_asynccnt(0);
#else
    asm volatile("s_wait_asynccnt 0x0" ::: "memory");
#endif
}

__global__ __launch_bounds__(128) void fine_matching_kernel(
    const float* __restrict__ feat0, const float* __restrict__ feat1,
    const float* __restrict__ mk0,   const float* __restrict__ mk1,
    const float* __restrict__ ex0,   const float* __restrict__ ex1,
    float* __restrict__ out, int M, int Nm)
{
    __shared__ __align__(16) float sF0[NL * FS];
    __shared__ __align__(16) float sF1[NR * FS];
    __shared__ float sC[NL * CS];
    __shared__ float sRowMax[NL], sRowSum[NL];
    __shared__ float sColMax[NR], sColSum[NR];
    __shared__ float sBestV[128];
    __shared__ int   sBestI[128];
    __shared__ float sWin[18];
    __shared__ int   sIdx[2];

    const int m    = blockIdx.x;
    const int tid  = threadIdx.x;
    const int lane = tid & 31;
    const int wv   = tid >> 5;

    // ---- stage features into LDS via CDNA5 async global->LDS copies (ASYNCcnt) ----
    {
        const char* g0 = (const char*)(feat0 + (size_t)m * NL * CCH);
        #pragma unroll
        for (int c = tid; c < NL * CCH / 4; c += 128) {
            int e = c << 2; int row = e >> 7; int col = e & 127;
            __builtin_amdgcn_global_load_async_to_lds_b128(
                (g_int4*)(g0 + (size_t)c * 16),
                (l_int4*)&sF0[row * FS + col], 0, 0);
        }
        const char* g1 = (const char*)(feat1 + (size_t)m * NR * CCH);
        for (int c = tid; c < NR * CCH / 4; c += 128) {
            int e = c << 2; int row = e >> 7; int col = e & 127;
            __builtin_amdgcn_global_load_async_to_lds_b128(
                (g_int4*)(g1 + (size_t)c * 16),
                (l_int4*)&sF1[row * FS + col], 0, 0);
        }
    }
    wait_asynccnt0();
    __syncthreads();

    // ---- conf_f = (f0[:, :120] @ f1[:, :120]^T) * (1/128) via V_WMMA_F32_16X16X4_F32 ----
    // wave wv owns l rows [16*wv, 16*wv+16); each wave does all 7 N tiles.
    {
        v8f acc[NTILE];
        #pragma unroll
        for (int n = 0; n < NTILE; ++n)
            #pragma unroll
            for (int v = 0; v < 8; ++v) acc[n][v] = 0.0f;

        const int hrow  = lane & 15;          // M (or N) index within tile
        const int khalf = (lane >> 4) << 1;   // lanes 16-31 hold K+2, K+3
        const int arow  = (wv << 4) + hrow;

        for (int k0 = 0; k0 < CFINE; k0 += 4) {
            v2f a;
            a.x = sF0[arow * FS + k0 + khalf];
            a.y = sF0[arow * FS + k0 + khalf + 1];
            #pragma unroll
            for (int n = 0; n < NTILE; ++n) {
                int col  = (n << 4) + hrow;
                int colc = col < NR ? col : NR - 1;   // padded cols: garbage, discarded
                v2f b;
                b.x = sF1[colc * FS + k0 + khalf];
                b.y = sF1[colc * FS + k0 + khalf + 1];
                acc[n] = __builtin_amdgcn_wmma_f32_16x16x4_f32(
                    false, a, false, b, (short)0, acc[n], false, false);
            }
        }

        const float sc = 1.0f / 128.0f;   // inv_c^2, C = 128
        #pragma unroll
        for (int n = 0; n < NTILE; ++n) {
            int col = (n << 4) + hrow;
            if (col < NR) {
                #pragma unroll
                for (int v = 0; v < 8; ++v) {
                    int row = (wv << 4) + v + ((lane >> 4) << 3); // C/D layout: lanes16-31 -> M+8
                    sC[row * CS + col] = acc[n][v] * sc;
                }
            }
        }
    }
    __syncthreads();

    // ---- dual-softmax stats ----
    if (tid < NL) {   // per-l stats over 100 r's
        float mx = -INFINITY;
        for (int r = 0; r < NR; ++r) mx = fmaxf(mx, sC[tid * CS + r]);
        float sum = 0.0f;
        for (int r = 0; r < NR; ++r) sum += expf(sC[tid * CS + r] - mx);
        sRowMax[tid] = mx; sRowSum[tid] = sum;
    }
    if (tid < NR) {   // per-r stats over 64 l's
        float mx = -INFINITY;
        for (int l = 0; l < NL; ++l) mx = fmaxf(mx, sC[l * CS + tid]);
        float sum = 0.0f;
        for (int l = 0; l < NL; ++l) sum += expf(sC[l * CS + tid] - mx);
        sColMax[tid] = mx; sColSum[tid] = sum;
    }
    __syncthreads();

    // ---- argmax over (l, inner 8x8 of r): 4096 candidates, 32 per thread ----
    // contiguous chunks + strictly-greater compares == JAX first-index tie-break
    {
        float bv = -INFINITY; int bi = 0;
        const int fbase = tid * 32;
        for (int s = 0; s < 32; ++s) {
            int f  = fbase + s;
            int l  = f >> 6;
            int ri = f & 63;
            int rf = ((ri >> 3) + 1) * WPAD + (ri & 7) + 1;  // inner (i,j) -> full 10x10 index
            float c = sC[l * CS + rf];
            float score = expf(2.0f * c - sRowMax[l] - sColMax[rf])
                          / (sRowSum[l] * sColSum[rf]);
            if (score > bv) { bv = score; bi = f; }
        }
        sBestV[tid] = bv; sBestI[tid] = bi;
    }
    __syncthreads();
    if (tid == 0) {
        float bv = sBestV[0]; int bi = sBestI[0];
        for (int t = 1; t < 128; ++t)
            if (sBestV[t] > bv) { bv = sBestV[t]; bi = sBestI[t]; }
        sIdx[0] = bi >> 6;   // idx_l
        sIdx[1] = bi & 63;   // idx_r
    }
    __syncthreads();

    // ---- 3x3 windows on last-8-channel correlations ----
    const float inv_s = 0.35355339059327373f;   // 1/sqrt(8)
    if (tid < 18) {
        int k  = tid % 9;
        int dy = k / 3 - 1, dx = k % 3 - 1;
        int il = sIdx[0], ir = sIdx[1];
        float v = 0.0f;
        if (tid < 9) {
            // cff[m, idx_l, (ir/8+dy)%10, (ir%8+dx)%10]
            int p = (ir / 8 + dy + WPAD) % WPAD;
            int q = (ir % 8 + dx + WPAD) % WPAD;
            int rfull = p * WPAD + q;
            #pragma unroll
            for (int c = 0; c < 8; ++c)
                v += sF0[il * FS + CFINE + c] * sF1[rfull * FS + CFINE + c];
        } else {
            // cffr is (M,100,64) reinterpreted as (M,64,10,10): flat f2 -> (l2, r2)
            int p = (il / 8 + dy + WPAD) % WPAD;
            int q = (il % 8 + dx + WPAD) % WPAD;
            int f2 = ir * NR + p * WPAD + q;
            int l2 = f2 >> 6, r2 = f2 & 63;
            #pragma unroll
            for (int c = 0; c < 8; ++c)
                v += sF1[l2 * FS + CFINE + c] * sF0[r2 * FS + CFINE + c];
        }
        sWin[tid] = v * inv_s;
    }
    __syncthreads();

    // ---- window softmax + spatial expectation + outputs ----
    if (tid == 0) {
        float cx[2], cy[2];   // [0] from cff -> coords_e ; [1] from cffr -> coords_er
        for (int g = 0; g < 2; ++g) {
            float mx = -INFINITY;
            for (int k = 0; k < 9; ++k) mx = fmaxf(mx, sWin[g * 9 + k] / 10.0f);
            float e[9], sum = 0.0f;
            for (int k = 0; k < 9; ++k) { e[k] = expf(sWin[g * 9 + k] / 10.0f - mx); sum += e[k]; }
            float sx = 0.0f, sy = 0.0f;
            for (int k = 0; k < 9; ++k) {
                float h = e[k] / sum;
                sx += h * (float)(k % 3 - 1);   // xs = [-1,0,1] over j
                sy += h * (float)(k / 3 - 1);   // ys = [-1,0,1] over i
            }
            cx[g] = sx; cy[g] = sy;
        }
        int il = sIdx[0], ir = sIdx[1];
        float glx = (float)(il % 8) - 3.5f, gly = (float)(il / 8) - 3.5f;
        float grx = (float)(ir % 8) - 3.5f, gry = (float)(ir / 8) - 3.5f;
        float d0x = (glx + cx[1]) * 8.0f;   // grid[idx_l] + coords_er, * SCALE
        float d0y = (gly + cy[1]) * 8.0f;
        float d1x = (grx + cx[0]) * 8.0f;   // grid[idx_r] + coords_e, * SCALE
        float d1y = (gry + cy[0]) * 8.0f;

        // out layout: [mkpts0_f (Nm,2) | mkpts1_f (Nm,2) | expec0 (M,2) | expec1 (M,2)]
        out[4 * Nm + 2 * m + 0]         = ex0[2 * m + 0] + d0x;
        out[4 * Nm + 2 * m + 1]         = ex0[2 * m + 1] + d0y;
        out[4 * Nm + 2 * M + 2 * m + 0] = ex1[2 * m + 0] + d1x;
        out[4 * Nm + 2 * M + 2 * m + 1] = ex1[2 * m + 1] + d1y;
        if (m < Nm) {
            out[2 * m + 0]          = mk0[2 * m + 0] + d0x;
            out[2 * m + 1]          = mk0[2 * m + 1] + d0y;
            out[2 * Nm + 2 * m + 0] = mk1[2 * m + 0] + d1x;
            out[2 * Nm + 2 * m + 1] = mk1[2 * m + 1] + d1y;
        }
    }
}

extern "C" void kernel_launch(void* const* d_in, const int* in_sizes, int n_in,
                              void* d_out, int out_size, void* d_ws, size_t ws_size,
                              hipStream_t stream) {
    const float* feat0 = (const float*)d_in[0];
    const float* feat1 = (const float*)d_in[1];
    const float* mk0   = (const float*)d_in[2];
    const float* mk1   = (const float*)d_in[3];
    const float* ex0   = (const float*)d_in[4];
    const float* ex1   = (const float*)d_in[5];
    float* out = (float*)d_out;

    const int M  = in_sizes[0] / (NL * CCH);
    const int Nm = in_sizes[2] / 2;

    fine_matching_kernel<<<M, 128, 0, stream>>>(feat0, feat1, mk0, mk1, ex0, ex1, out, M, Nm);
}